// SphericalExpansion_34127810134280
// MI455X (gfx1250) — compile-verified
//
#include <hip/hip_runtime.h>
#include <math.h>

typedef __attribute__((ext_vector_type(2))) float v2f;
typedef __attribute__((ext_vector_type(8))) float v8f;

#define PI_F 3.14159265358979323846f

// ---------------- Pass 1: histogram of center indices ----------------
__global__ void se_hist_kernel(const int* __restrict__ iidx, int* __restrict__ cnt, int n) {
    int p = blockIdx.x * blockDim.x + threadIdx.x;
    if (p < n) atomicAdd(&cnt[iidx[p]], 1);
}

// ---------------- Pass 2: exclusive scan (single block) ----------------
__global__ void se_scan_kernel(const int* __restrict__ cnt, int* __restrict__ offs,
                               int* __restrict__ cursor, int n) {
    __shared__ int buf[1024];
    __shared__ int carry;
    int t = threadIdx.x;
    if (t == 0) carry = 0;
    __syncthreads();
    for (int base = 0; base < n; base += 1024) {
        int idx = base + t;
        int v = (idx < n) ? cnt[idx] : 0;
        buf[t] = v;
        __syncthreads();
        for (int off = 1; off < 1024; off <<= 1) {
            int x = (t >= off) ? buf[t - off] : 0;
            __syncthreads();
            buf[t] += x;
            __syncthreads();
        }
        int excl = buf[t] - v;
        if (idx < n) { int o = carry + excl; offs[idx] = o; cursor[idx] = o; }
        __syncthreads();
        if (t == 0) carry += buf[1023];
        __syncthreads();
    }
    if (t == 0) offs[n] = carry;
}

// ---------------- Pass 3: scatter pair ids into segment order ----------------
__global__ void se_scatter_kernel(const int* __restrict__ iidx, int* __restrict__ cursor,
                                  int* __restrict__ order, int n) {
    int p = blockIdx.x * blockDim.x + threadIdx.x;
    if (p < n) {
        int pos = atomicAdd(&cursor[iidx[p]], 1);
        order[pos] = p;
    }
}

// ---------------- Pass 4: per-center WMMA accumulation ----------------
// One wave32 per center.  out[16 x 36] = sum_p Y[p][16] (x) F[p][36]
// computed as a [16 x K] * [K x 48] matmul via v_wmma_f32_16x16x4_f32,
// K = pairs in the segment, zero padded to multiples of 32.
__global__ __launch_bounds__(32)
void se_center_kernel(const float* __restrict__ R,
                      const float* __restrict__ W,
                      const int*  __restrict__ jidx,
                      const int*  __restrict__ species,
                      const int*  __restrict__ offs,
                      const int*  __restrict__ order,
                      float* __restrict__ out,
                      size_t b1, size_t b2, size_t b3) {
    __shared__ float Yst[32][16];   // per staged pair: 16 spherical harmonics
    __shared__ float Fst[32][48];   // per staged pair: rad (x) species, padded 36->48
    __shared__ float Dst[16][48];   // result dump

    const int a    = blockIdx.x;
    const int lane = threadIdx.x;
    const int half = lane >> 4;     // 0: lanes 0-15, 1: lanes 16-31
    const int lm   = lane & 15;

    const int start = offs[a];
    const int end   = offs[a + 1];

    v8f c0 = {}, c1 = {}, c2 = {};

    for (int base = start; base < end; base += 32) {
        int pi = base + lane;
        if (pi < end) {
            int p = order[pi];
            float x = R[3 * p + 0], y = R[3 * p + 1], z = R[3 * p + 2];
            float r   = sqrtf(x * x + y * y + z * z);
            float inv = 1.0f / r;
            float ux = x * inv, uy = y * inv, uz = z * inv;
            float x2 = ux * ux, y2 = uy * uy, z2 = uz * uz;

            float Yv[16];
            Yv[0]  = 0.28209479177387814f;
            Yv[1]  = 0.4886025119029199f * uy;
            Yv[2]  = 0.4886025119029199f * uz;
            Yv[3]  = 0.4886025119029199f * ux;
            Yv[4]  = 1.0925484305920792f * ux * uy;
            Yv[5]  = 1.0925484305920792f * uy * uz;
            Yv[6]  = 0.31539156525252005f * (3.0f * z2 - 1.0f);
            Yv[7]  = 1.0925484305920792f * ux * uz;
            Yv[8]  = 0.5462742152960396f * (x2 - y2);
            Yv[9]  = 0.5900435899266435f * uy * (3.0f * x2 - y2);
            Yv[10] = 2.890611442640554f  * ux * uy * uz;
            Yv[11] = 0.4570457994644658f * uy * (5.0f * z2 - 1.0f);
            Yv[12] = 0.3731763325901154f * uz * (5.0f * z2 - 3.0f);
            Yv[13] = 0.4570457994644658f * ux * (5.0f * z2 - 1.0f);
            Yv[14] = 1.445305721320277f  * uz * (x2 - y2);
            Yv[15] = 0.5900435899266435f * ux * (x2 - 3.0f * y2);
#pragma unroll
            for (int m = 0; m < 16; ++m) Yst[lane][m] = Yv[m];

            // radial basis * shifted-cosine cutoff
            float fcut;
            if (r < 4.5f)      fcut = 1.0f;
            else if (r < 5.0f) fcut = 0.5f * (1.0f + cosf(2.0f * PI_F * (r - 4.5f)));
            else               fcut = 0.0f;
            float pref = 0.6324555320336759f * inv * fcut;   // sqrt(2/CUTOFF)
            float radv[9];
#pragma unroll
            for (int n = 0; n < 9; ++n)
                radv[n] = pref * sinf((float)(n + 1) * (PI_F / 5.0f) * r);

            int s = species[jidx[p]];
            float sp0 = W[4 * s + 0], sp1 = W[4 * s + 1];
            float sp2 = W[4 * s + 2], sp3 = W[4 * s + 3];
#pragma unroll
            for (int n = 0; n < 9; ++n) {
                Fst[lane][4 * n + 0] = radv[n] * sp0;
                Fst[lane][4 * n + 1] = radv[n] * sp1;
                Fst[lane][4 * n + 2] = radv[n] * sp2;
                Fst[lane][4 * n + 3] = radv[n] * sp3;
            }
#pragma unroll
            for (int q = 36; q < 48; ++q) Fst[lane][q] = 0.0f;
        } else {
#pragma unroll
            for (int m = 0; m < 16; ++m) Yst[lane][m] = 0.0f;
#pragma unroll
            for (int q = 0; q < 48; ++q) Fst[lane][q] = 0.0f;
        }
        __syncthreads();

        // A[M,K] = Yst[k+K][M], B[K,N] = Fst[k+K][N]
        // f32 16x4 A layout: v0 = K{0,2}, v1 = K{1,3} (halves of the wave)
#pragma unroll
        for (int k = 0; k < 32; k += 4) {
            v2f A;
            A.x = Yst[k + 2 * half + 0][lm];
            A.y = Yst[k + 2 * half + 1][lm];
            v2f B0, B1, B2;
            B0.x = Fst[k + 2 * half + 0][lm];
            B0.y = Fst[k + 2 * half + 1][lm];
            B1.x = Fst[k + 2 * half + 0][16 + lm];
            B1.y = Fst[k + 2 * half + 1][16 + lm];
            B2.x = Fst[k + 2 * half + 0][32 + lm];
            B2.y = Fst[k + 2 * half + 1][32 + lm];
            c0 = __builtin_amdgcn_wmma_f32_16x16x4_f32(false, A, false, B0, (short)0, c0, false, false);
            c1 = __builtin_amdgcn_wmma_f32_16x16x4_f32(false, A, false, B1, (short)0, c1, false, false);
            c2 = __builtin_amdgcn_wmma_f32_16x16x4_f32(false, A, false, B2, (short)0, c2, false, false);
        }
        __syncthreads();
    }

    // D layout: lanes 0-15 vgpr v -> C[M=v][N=lane]; lanes 16-31 -> C[M=v+8][N=lane-16]
#pragma unroll
    for (int v = 0; v < 8; ++v) {
        int row = v + 8 * half;
        Dst[row][lm]      = c0[v];
        Dst[row][16 + lm] = c1[v];
        Dst[row][32 + lm] = c2[v];
    }
    __syncthreads();

    // scatter the 16x36 tile into the four l-blocks of the concatenated output
    for (int idx = lane; idx < 576; idx += 32) {
        int m = idx / 36;
        int q = idx - m * 36;
        float val = Dst[m][q];
        size_t addr;
        if (m == 0)      addr =      (size_t)a * 36  + (size_t)q;
        else if (m < 4)  addr = b1 + (size_t)a * 108 + (size_t)(m - 1) * 36 + (size_t)q;
        else if (m < 9)  addr = b2 + (size_t)a * 180 + (size_t)(m - 4) * 36 + (size_t)q;
        else             addr = b3 + (size_t)a * 252 + (size_t)(m - 9) * 36 + (size_t)q;
        out[addr] = val;
    }
}

extern "C" void kernel_launch(void* const* d_in, const int* in_sizes, int n_in,
                              void* d_out, int out_size, void* d_ws, size_t ws_size,
                              hipStream_t stream) {
    const float* R       = (const float*)d_in[0];   // [P,3]
    const float* W       = (const float*)d_in[1];   // [S,4]
    const int*   iidx    = (const int*)d_in[2];     // [P]
    const int*   jidx    = (const int*)d_in[3];     // [P]
    const int*   species = (const int*)d_in[4];     // [A]

    int n_pairs = in_sizes[2];
    int n_atoms = in_sizes[4];

    int* ws     = (int*)d_ws;
    int* cnt    = ws;                         // [A]
    int* offs   = cnt + n_atoms;              // [A+1]
    int* cursor = offs + n_atoms + 8;         // [A]
    int* order  = cursor + n_atoms;           // [P]

    hipMemsetAsync(cnt, 0, (size_t)n_atoms * sizeof(int), stream);

    const int tb = 256;
    se_hist_kernel<<<(n_pairs + tb - 1) / tb, tb, 0, stream>>>(iidx, cnt, n_pairs);
    se_scan_kernel<<<1, 1024, 0, stream>>>(cnt, offs, cursor, n_atoms);
    se_scatter_kernel<<<(n_pairs + tb - 1) / tb, tb, 0, stream>>>(iidx, cursor, order, n_pairs);

    size_t b1 = (size_t)n_atoms * 36;
    size_t b2 = b1 + (size_t)n_atoms * 108;
    size_t b3 = b2 + (size_t)n_atoms * 180;
    se_center_kernel<<<n_atoms, 32, 0, stream>>>(R, W, jidx, species, offs, order,
                                                 (float*)d_out, b1, b2, b3);
}